// MHPLSTMCore_88235808129369
// MI455X (gfx1250) — compile-verified
//
#include <hip/hip_runtime.h>
#include <hip/hip_bf16.h>
#include <stdint.h>

typedef __attribute__((ext_vector_type(2))) float v2f;
typedef __attribute__((ext_vector_type(8))) float v8f;

#define BB 4
#define SS 4096
#define HH 8
#define DD 64
#define HD 512            // H*D
#define CS 256            // chunk size along S for scans
#define NC 16             // SS/CS
#define MTOT (BB*SS*HH)   // 131072 GEMM rows
#define KDIM 128          // 2*D
#define MPANEL 64         // rows per block in GEMM
#define PAD 132           // LDS row pad (floats): 528B row = 33*16B -> b128-aligned, conflict-free

__device__ __forceinline__ float sigmoidf_(float x) {
    return 1.0f / (1.0f + __expf(-x));
}

// ---- CDNA5 async global->LDS copies (ASYNCcnt-tracked, bypass VGPRs) ----
__device__ __forceinline__ void async_ldst_b128(uint32_t lds_off, const void* g) {
    asm volatile("global_load_async_to_lds_b128 %0, %1, off"
                 :: "v"(lds_off), "v"((uint64_t)(uintptr_t)g) : "memory");
}
__device__ __forceinline__ void async_ldst_b32(uint32_t lds_off, const void* g) {
    asm volatile("global_load_async_to_lds_b32 %0, %1, off"
                 :: "v"(lds_off), "v"((uint64_t)(uintptr_t)g) : "memory");
}
__device__ __forceinline__ void wait_async0() {
    asm volatile("s_wait_asynccnt 0x0" ::: "memory");
}
// low 32 bits of a generic pointer into LDS == LDS byte offset (flat aperture rule)
__device__ __forceinline__ uint32_t lds_off32(const void* p) {
    return (uint32_t)(uintptr_t)p;
}

// ---------------- Pass 1: per-chunk column sums of heads_input ----------------
__global__ __launch_bounds__(HD) void k_chunk_sum(const float* __restrict__ x,
                                                  float* __restrict__ sums) {
    const int b = blockIdx.x >> 4, c = blockIdx.x & 15;
    const int col = threadIdx.x;
    const int base = (b * SS + c * CS) * HD + col;
    float s = 0.0f;
    for (int i = 0; i < CS; ++i) {
        s += x[base + i * HD];
    }
    sums[blockIdx.x * HD + col] = s;
}

// ---------------- Pass 2: exclusive scan of chunk sums (per b,col) ----------------
__global__ __launch_bounds__(HD) void k_scan_sums(float* __restrict__ sums) {
    const int b = blockIdx.x;
    const int col = threadIdx.x;
    float run = 0.0f;
    for (int c = 0; c < NC; ++c) {
        const int i = (b * NC + c) * HD + col;
        const float t = sums[i];
        sums[i] = run;
        run += t;
    }
}

// ------ Pass 3: in-chunk exclusive cumsum + LayerNorm over (H,D) per token ------
__global__ __launch_bounds__(HD) void k_cumsum_ln(const float* __restrict__ x,
                                                  const float* __restrict__ offs,
                                                  const float* __restrict__ gamma,
                                                  const float* __restrict__ beta,
                                                  float* __restrict__ csum_ln) {
    __shared__ float red1[16];
    __shared__ float red2[16];
    __shared__ float mv[2];
    const int b = blockIdx.x >> 4, c = blockIdx.x & 15;
    const int col = threadIdx.x;
    const int lane = col & 31, wid = col >> 5;
    float run = offs[blockIdx.x * HD + col];
    const float g = gamma[col], be = beta[col];
    const int base = (b * SS + c * CS) * HD + col;
    for (int s = 0; s < CS; ++s) {
        const float val = run;               // exclusive cumsum value
        run += x[base + s * HD];
        float s1 = val, s2 = val * val;
        #pragma unroll
        for (int off = 16; off; off >>= 1) {
            s1 += __shfl_xor(s1, off, 32);
            s2 += __shfl_xor(s2, off, 32);
        }
        if (lane == 0) { red1[wid] = s1; red2[wid] = s2; }
        __syncthreads();
        if (col < 16) {
            float t1 = red1[col], t2 = red2[col];
            #pragma unroll
            for (int off = 8; off; off >>= 1) {
                t1 += __shfl_xor(t1, off, 32);
                t2 += __shfl_xor(t2, off, 32);
            }
            if (col == 0) { mv[0] = t1; mv[1] = t2; }
        }
        __syncthreads();
        const float mean = mv[0] * (1.0f / 512.0f);
        const float var  = mv[1] * (1.0f / 512.0f) - mean * mean;
        const float inv  = rsqrtf(var + 1e-5f);
        csum_ln[base + s * HD] = (val - mean) * inv * g + be;
    }
}

// ---------------- GEMM1: [M,128]x[128,192] + gates, f32 WMMA ----------------
__global__ __launch_bounds__(128) void k_gemm_hid(const float* __restrict__ x,
                                                  const float* __restrict__ csum,
                                                  const float* __restrict__ W,
                                                  const float* __restrict__ bias,
                                                  float* __restrict__ fg,
                                                  float* __restrict__ igh) {
    __shared__ float Asm[MPANEL * PAD];   // 64x128 A-panel (padded)
    __shared__ float Bsm[16 * PAD];       // 16x128 W^T tile (padded)
    const int tid = threadIdx.x;
    const int rowBase = blockIdx.x * MPANEL;

    // Stage A panel via async b128 DMA: cols [0,64)=heads_input, [64,128)=csum_ln
    for (int e = tid; e < MPANEL * 32; e += 128) {
        const int r = e >> 5, c4 = e & 31;
        const float* src = (c4 < 16) ? (x + (rowBase + r) * DD + c4 * 4)
                                     : (csum + (rowBase + r) * DD + (c4 - 16) * 4);
        async_ldst_b128(lds_off32(&Asm[r * PAD + c4 * 4]), src);
    }
    wait_async0();
    __syncthreads();

    const int lane = tid & 31;
    const int wave = tid >> 5;                 // 0..3 : 16-row M tile each
    const int mloc = wave * 16 + (lane & 15);
    const int dsel = (lane >> 4) << 1;         // K offset 0 or 2 per lane half
    const int nl = lane & 15;
    const int rowHalf = (lane >> 4) << 3;      // +8 rows for lanes 16..31

    // Preload all A fragments (32 k-steps x v2f = 64 VGPRs)
    v2f afrag[32];
    #pragma unroll
    for (int kt = 0; kt < 32; ++kt)
        afrag[kt] = *(const v2f*)&Asm[mloc * PAD + kt * 4 + dsel];

    float isig[4][8];   // saved sigmoid(i-gate) tiles, kept in regs via full unroll

    #pragma unroll
    for (int t = 0; t < 12; ++t) {            // N tiles: 0-3=i, 4-7=f, 8-11=h
        __syncthreads();
        // transpose-stage W tile via per-element async b32 DMA
        for (int e = tid; e < 16 * KDIM; e += 128) {
            const int n = e & 15, k = e >> 4;
            async_ldst_b32(lds_off32(&Bsm[n * PAD + k]), W + k * 192 + t * 16 + n);
        }
        wait_async0();
        __syncthreads();
        v8f acc = {};
        #pragma unroll
        for (int kt = 0; kt < 32; ++kt) {
            const v2f bfrag = *(const v2f*)&Bsm[nl * PAD + kt * 4 + dsel];
            acc = __builtin_amdgcn_wmma_f32_16x16x4_f32(
                false, afrag[kt], false, bfrag, (short)0, acc, false, false);
        }
        const float bv = bias[t * 16 + nl];
        const int dcol = (t & 3) * 16 + nl;
        if (t < 4) {
            #pragma unroll
            for (int cc = 0; cc < 8; ++cc)
                isig[t][cc] = sigmoidf_(acc[cc] + bv);
        } else if (t < 8) {
            #pragma unroll
            for (int cc = 0; cc < 8; ++cc) {
                const int r = rowBase + wave * 16 + rowHalf + cc;
                fg[r * DD + dcol] = sigmoidf_(acc[cc] + bv);
            }
        } else {
            #pragma unroll
            for (int cc = 0; cc < 8; ++cc) {
                const int r = rowBase + wave * 16 + rowHalf + cc;
                const float h = acc[cc] + bv;
                igh[r * DD + dcol] = isig[t - 8][cc] * fmaxf(h, 0.0f);
            }
        }
    }
}

// -------- cell scan pass 1: per-chunk (A,B) composition of c <- f*c + g --------
__global__ __launch_bounds__(HD) void k_gate_chunk(const float* __restrict__ fg,
                                                   const float* __restrict__ igh,
                                                   float* __restrict__ Ac,
                                                   float* __restrict__ Bc) {
    const int b = blockIdx.x >> 4, c = blockIdx.x & 15;
    const int col = threadIdx.x;
    const int base = (b * SS + c * CS) * HD + col;
    float A = 1.0f, Bv = 0.0f;
    for (int s = 0; s < CS; ++s) {
        const float f = fg[base + s * HD];
        const float g = igh[base + s * HD];
        A *= f;
        Bv = Bv * f + g;
    }
    Ac[blockIdx.x * HD + col] = A;
    Bc[blockIdx.x * HD + col] = Bv;
}

// -------- cell scan pass 2: cross-chunk carry of cell start values --------
__global__ __launch_bounds__(HD) void k_scan_cell(const float* __restrict__ Ac,
                                                  const float* __restrict__ Bc,
                                                  const float* __restrict__ init_cx,
                                                  float* __restrict__ carry) {
    const int b = blockIdx.x;
    const int col = threadIdx.x;
    float start = init_cx[col];       // cell_{-1}
    for (int c = 0; c < NC; ++c) {
        const int i = (b * NC + c) * HD + col;
        carry[i] = start;
        start = Ac[i] * start + Bc[i];
    }
}

// -------- cell scan pass 3: apply recurrence within chunk --------
__global__ __launch_bounds__(HD) void k_cell_apply(const float* __restrict__ fg,
                                                   const float* __restrict__ igh,
                                                   const float* __restrict__ carry,
                                                   float* __restrict__ cell) {
    const int b = blockIdx.x >> 4, c = blockIdx.x & 15;
    const int col = threadIdx.x;
    const int base = (b * SS + c * CS) * HD + col;
    float cprev = carry[blockIdx.x * HD + col];
    for (int s = 0; s < CS; ++s) {
        const float cv = fg[base + s * HD] * cprev + igh[base + s * HD];
        cell[base + s * HD] = cv;
        cprev = cv;
    }
}

// ---------------- GEMM2: [M,128]x[128,64] + sigmoid, out = og*cell ----------------
__global__ __launch_bounds__(128) void k_gemm_og(const float* __restrict__ x,
                                                 const float* __restrict__ cell,
                                                 const float* __restrict__ W,
                                                 const float* __restrict__ bias,
                                                 float* __restrict__ out) {
    __shared__ float Asm[MPANEL * PAD];
    __shared__ float Bsm[16 * PAD];
    const int tid = threadIdx.x;
    const int rowBase = blockIdx.x * MPANEL;

    for (int e = tid; e < MPANEL * 32; e += 128) {
        const int r = e >> 5, c4 = e & 31;
        const float* src = (c4 < 16) ? (x + (rowBase + r) * DD + c4 * 4)
                                     : (cell + (rowBase + r) * DD + (c4 - 16) * 4);
        async_ldst_b128(lds_off32(&Asm[r * PAD + c4 * 4]), src);
    }
    wait_async0();
    __syncthreads();

    const int lane = tid & 31;
    const int wave = tid >> 5;
    const int mloc = wave * 16 + (lane & 15);
    const int dsel = (lane >> 4) << 1;
    const int nl = lane & 15;
    const int rowHalf = (lane >> 4) << 3;

    v2f afrag[32];
    #pragma unroll
    for (int kt = 0; kt < 32; ++kt)
        afrag[kt] = *(const v2f*)&Asm[mloc * PAD + kt * 4 + dsel];

    #pragma unroll
    for (int t = 0; t < 4; ++t) {             // N = 64 -> 4 tiles
        __syncthreads();
        for (int e = tid; e < 16 * KDIM; e += 128) {
            const int n = e & 15, k = e >> 4;
            async_ldst_b32(lds_off32(&Bsm[n * PAD + k]), W + k * DD + t * 16 + n);
        }
        wait_async0();
        __syncthreads();
        v8f acc = {};
        #pragma unroll
        for (int kt = 0; kt < 32; ++kt) {
            const v2f bfrag = *(const v2f*)&Bsm[nl * PAD + kt * 4 + dsel];
            acc = __builtin_amdgcn_wmma_f32_16x16x4_f32(
                false, afrag[kt], false, bfrag, (short)0, acc, false, false);
        }
        const float bv = bias[t * 16 + nl];
        const int dcol = t * 16 + nl;
        #pragma unroll
        for (int cc = 0; cc < 8; ++cc) {
            const int r = rowBase + wave * 16 + rowHalf + cc;
            const float cv = cell[r * DD + dcol];
            out[r * DD + dcol] = sigmoidf_(acc[cc] + bv) * cv;
        }
    }
}

extern "C" void kernel_launch(void* const* d_in, const int* in_sizes, int n_in,
                              void* d_out, int out_size, void* d_ws, size_t ws_size,
                              hipStream_t stream) {
    const float* x       = (const float*)d_in[0];  // [B,S,H,D]
    const float* W_hid   = (const float*)d_in[1];  // [128,192]
    const float* b_hid   = (const float*)d_in[2];  // [192]
    const float* W_og    = (const float*)d_in[3];  // [128,64]
    const float* b_og    = (const float*)d_in[4];  // [64]
    const float* gamma   = (const float*)d_in[5];  // [H,D]=512
    const float* beta    = (const float*)d_in[6];  // [512]
    const float* init_cx = (const float*)d_in[7];  // [512]
    float* out = (float*)d_out;
    float* ws  = (float*)d_ws;

    // workspace layout (floats)
    const size_t O_SUMS  = 0;                         // B*NC*HD = 32768
    const size_t O_AC    = O_SUMS  + (size_t)BB * NC * HD;
    const size_t O_BC    = O_AC    + (size_t)BB * NC * HD;
    const size_t O_CARRY = O_BC    + (size_t)BB * NC * HD;
    const size_t O_CSUM  = O_CARRY + (size_t)BB * NC * HD;   // 8388608, reused as cell
    const size_t O_FG    = O_CSUM  + (size_t)MTOT * DD;
    const size_t O_IGH   = O_FG    + (size_t)MTOT * DD;

    float* sums  = ws + O_SUMS;
    float* Ac    = ws + O_AC;
    float* Bc    = ws + O_BC;
    float* carry = ws + O_CARRY;
    float* csum  = ws + O_CSUM;   // layernormed exclusive cumsum
    float* cellb = ws + O_CSUM;   // reuse after GEMM1 consumed csum
    float* fg    = ws + O_FG;
    float* igh   = ws + O_IGH;

    k_chunk_sum <<<BB * NC, HD, 0, stream>>>(x, sums);
    k_scan_sums <<<BB,      HD, 0, stream>>>(sums);
    k_cumsum_ln <<<BB * NC, HD, 0, stream>>>(x, sums, gamma, beta, csum);
    k_gemm_hid  <<<MTOT / MPANEL, 128, 0, stream>>>(x, csum, W_hid, b_hid, fg, igh);
    k_gate_chunk<<<BB * NC, HD, 0, stream>>>(fg, igh, Ac, Bc);
    k_scan_cell <<<BB,      HD, 0, stream>>>(Ac, Bc, init_cx, carry);
    k_cell_apply<<<BB * NC, HD, 0, stream>>>(fg, igh, carry, cellb);
    k_gemm_og   <<<MTOT / MPANEL, 128, 0, stream>>>(x, cellb, W_og, b_og, out);
}